// Hbv_2_5214090298013
// MI455X (gfx1250) — compile-verified
//
#include <hip/hip_runtime.h>
#include <stdint.h>

// HBV-2.0 fused scan + gamma-UH routing for MI455X (gfx1250, wave32).
// Memory-bound streaming scan (~234 MB @ 23.3 TB/s ~ 10us floor):
//  - TDM tensor_load_to_lds for the static-parameter block gather
//  - double-buffered global_load_async_to_lds_b64/b128 for per-timestep data
//  - all state in registers, NMUL-mean via wave32 lane-pair shuffle,
//    15-tap routing fused on a register ring buffer (Qsim never hits HBM).

#define T_STEPS  730
#define N_GRID   10000
#define GPB      128            // grid cells per block
#define BLOCK    256            // 2 lanes per cell (NMUL components)
#define CHUNK    8              // timesteps staged per async batch
#define LENF     15
#define NEARZERO 1e-5f

typedef uint32_t v4u __attribute__((ext_vector_type(4)));
typedef int      v8i __attribute__((ext_vector_type(8)));
typedef int      v4i __attribute__((ext_vector_type(4)));

// ---- CDNA5 async global -> LDS copies (GV mode: 64-bit vaddr pair, saddr=off).
__device__ __forceinline__ void async_copy_b64(uint32_t lds_off, const float* gptr) {
    asm volatile("global_load_async_to_lds_b64 %0, %1, off"
                 :: "v"(lds_off), "v"((uint64_t)(uintptr_t)gptr) : "memory");
}
__device__ __forceinline__ void async_copy_b128(uint32_t lds_off, const float* gptr) {
    asm volatile("global_load_async_to_lds_b128 %0, %1, off"
                 :: "v"(lds_off), "v"((uint64_t)(uintptr_t)gptr) : "memory");
}
__device__ __forceinline__ void wait_async0() {
    asm volatile("s_wait_asynccnt 0x0" ::: "memory");
}

__global__ __launch_bounds__(BLOCK) void hbv_fused_kernel(
    const float* __restrict__ x_phy,       // [T, G, 3]  P, Tair, PET
    const float* __restrict__ ac_all,      // [G]
    const float* __restrict__ params_dy,   // [T, G, 2, NMUL] -> 4 floats
    const float* __restrict__ params_stat, // [G, 30]
    float* __restrict__ out)               // [T, G]
{
    // 2*CHUNK*384 (x) + 2*CHUNK*512 (dy) floats = 57344 B; the first 3840
    // floats double as the TDM staging area for params_stat during init.
    __shared__ __align__(16) float smem[2 * CHUNK * GPB * 3 + 2 * CHUNK * GPB * 4];
    float* s_xb = smem;                        // [2][CHUNK][GPB*3]
    float* s_db = smem + 2 * CHUNK * GPB * 3;  // [2][CHUNK][GPB*4]
#define SX(b, tt, e) s_xb[(((b) * CHUNK) + (tt)) * (GPB * 3) + (e)]
#define SD(b, tt, e) s_db[(((b) * CHUNK) + (tt)) * (GPB * 4) + (e)]
    float* s_stat = smem;                      // GPB*30 = 3840 floats (init only)

    const int tid = threadIdx.x;
    const int gl  = tid >> 1;        // local cell 0..127
    const int m   = tid & 1;         // NMUL component
    const int g0  = blockIdx.x * GPB;
    const int g   = g0 + gl;
    const bool writer = (m == 0) && (g < N_GRID);
    const int gc  = (g < N_GRID) ? g : (N_GRID - 1);
    const int cells = (N_GRID - g0) < GPB ? (N_GRID - g0) : GPB;

    // ================= TDM: stage params_stat[g0 : g0+cells, 0:30] ===========
    // One contiguous span of cells*30 floats; tile_dim0 = GPB*30, tensor_dim0
    // clamped to the valid tail so the OOB remainder zero-fills deterministically.
    if (tid < 32) {  // wave 0 only (TDM ignores EXEC; per-wave instruction)
        const uint64_t ga  = (uint64_t)(uintptr_t)(params_stat + (size_t)g0 * 30);
        const uint32_t lds = (uint32_t)(uintptr_t)s_stat;
        const uint32_t td0 = (uint32_t)(cells * 30);        // valid elements
        const uint32_t tl0 = (uint32_t)(GPB * 30);          // tile dim 0 (<=65535)
        v4u d0 = { 1u,                                      // count=1
                   lds,                                     // lds_addr [63:32]
                   (uint32_t)ga,                            // global_addr lo
                   (uint32_t)((ga >> 32) & 0x01FFFFFFu) | (2u << 30) }; // ga hi | type=2
        v8i d1 = { (int)(2u << 16),                         // data_size=4B, mask=0
                   (int)((td0 & 0xFFFFu) << 16),            // tensor_dim0 lo
                   (int)((td0 >> 16) | (1u << 16)),         // tensor_dim0 hi | tensor_dim1=1
                   (int)(tl0 << 16),                        // tile_dim0
                   (int)1,                                  // tile_dim1=1, tile_dim2=0
                   (int)tl0,                                // tensor_dim0_stride lo
                   0, 0 };
        v4i dz = { 0, 0, 0, 0 };
#if defined(__clang_major__) && (__clang_major__ >= 23)
        v8i dz8 = { 0, 0, 0, 0, 0, 0, 0, 0 };
        __builtin_amdgcn_tensor_load_to_lds(d0, d1, dz, dz, dz8, 0);
#else
        __builtin_amdgcn_tensor_load_to_lds(d0, d1, dz, dz, 0);
#endif
        __builtin_amdgcn_s_wait_tensorcnt(0);
    }
    __syncthreads();

    // ---------------- static params from LDS, scaled to physical bounds ------
    const float* ps = s_stat + gl * 30;
    const float parFC    = ps[0*2+m]  * 950.0f   + 50.0f;
    const float parK0    = ps[1*2+m]  * 0.85f    + 0.05f;
    const float parK1    = ps[2*2+m]  * 0.49f    + 0.01f;
    const float parK2    = ps[3*2+m]  * 0.199f   + 0.001f;
    const float parLP    = ps[4*2+m]  * 0.8f     + 0.2f;
    const float parPERC  = ps[5*2+m]  * 10.0f;
    const float parUZL   = ps[6*2+m]  * 100.0f;
    const float parTT    = ps[7*2+m]  * 5.0f     - 2.5f;
    const float parCFMAX = ps[8*2+m]  * 9.5f     + 0.5f;
    const float parCFR   = ps[9*2+m]  * 0.1f;
    const float parCWH   = ps[10*2+m] * 0.2f;
    const float parC     = ps[11*2+m];
    const float parRT    = ps[12*2+m] * 20.0f;
    const float parAC    = ps[13*2+m] * 2500.0f;
    const float Ac       = ac_all[gc];

    const float gwmax   = parRT * fminf(fmaxf(1.0f - Ac / (parAC + NEARZERO), 0.0f), 1.0f);
    const float refrC   = parCFR * parCFMAX;
    const float invFC   = 1.0f / parFC;
    const float invLPFC = 1.0f / (parLP * parFC);

    // ---- gamma-UH weights: w_k ∝ t_k^(a-1) exp(-t_k/θ); the
    // exp(-lgamma(a))·θ^-a prefactor cancels under normalization, and
    // ln(t_k) are compile-time constants — no powf/lgammaf needed.
    const float a  = fmaxf(ps[28] * 2.9f, 0.0f) + 0.1f;
    const float th = fmaxf(ps[29] * 6.5f, 0.0f) + 0.5f;
    const float LOGTK[LENF] = {
        -0.69314718f, 0.40546511f, 0.91629073f, 1.25276297f, 1.50407740f,
         1.70474809f, 1.87180218f, 2.01490302f, 2.14006616f, 2.25129180f,
         2.35137526f, 2.44234704f, 2.52572864f, 2.60268969f, 2.67414865f };
    const float am1 = a - 1.0f;
    const float nith = -1.0f / th;
    float w[LENF];
    float wsum = 0.0f;
    #pragma unroll
    for (int k = 0; k < LENF; ++k) {
        const float tk = (float)k + 0.5f;
        w[k] = __expf(am1 * LOGTK[k] + tk * nith);
        wsum += w[k];
    }
    const float winv = 1.0f / wsum;
    #pragma unroll
    for (int k = 0; k < LENF; ++k) w[k] *= winv;

    // ---------------- state (registers) ----------------
    float SNOWPACK = 1e-3f, MELTWATER = 1e-3f, SM = 1e-3f, SUZ = 1e-3f, SLZ = 1e-3f;
    float h[LENF - 1];
    #pragma unroll
    for (int i = 0; i < LENF - 1; ++i) h[i] = 0.0f;

    __syncthreads();   // everyone done reading s_stat before overlay reuse

    // x rows start 8B-aligned ((even)*12 B), dy rows 16B-aligned ((int)*16 B)
    const int nx64  = (cells * 3) >> 1;   // B64 transfers per x row
    const int nd128 = cells;              // B128 transfers per dy row

    auto stage = [&](int bufi, int cbase) {
        for (int tt = 0; tt < CHUNK; ++tt) {
            const int t = cbase + tt;
            if (t >= T_STEPS) break;
            const float* gx = x_phy + ((size_t)t * N_GRID + g0) * 3;
            for (int e = tid; e < nx64; e += BLOCK)
                async_copy_b64((uint32_t)(uintptr_t)&SX(bufi, tt, e * 2), gx + e * 2);
            const float* gd = params_dy + ((size_t)t * N_GRID + g0) * 4;
            for (int e = tid; e < nd128; e += BLOCK)
                async_copy_b128((uint32_t)(uintptr_t)&SD(bufi, tt, e * 4), gd + e * 4);
        }
    };

    stage(0, 0);
    wait_async0();
    __syncthreads();

    int buf = 0;
    for (int cb = 0; cb < T_STEPS; cb += CHUNK) {
        if (cb + CHUNK < T_STEPS) stage(buf ^ 1, cb + CHUNK);   // prefetch next chunk
        const int lim = (T_STEPS - cb) < CHUNK ? (T_STEPS - cb) : CHUNK;
        for (int tt = 0; tt < lim; ++tt) {
            const float Pt   = SX(buf, tt, gl * 3 + 0);
            const float Tt   = SX(buf, tt, gl * 3 + 1);
            const float PETt = SX(buf, tt, gl * 3 + 2);
            const float beta   = SD(buf, tt, gl * 4 + m)     * 5.0f + 1.0f;   // parBETA
            const float betaet = SD(buf, tt, gl * 4 + 2 + m) * 4.7f + 0.3f;   // parBETAET

            // ---- snow bucket ----
            const float rain_mask = (Tt >= parTT) ? 1.0f : 0.0f;
            const float RAIN = Pt * rain_mask;
            const float SNOW = Pt - RAIN;
            SNOWPACK += SNOW;
            const float melt = fminf(fmaxf(parCFMAX * (Tt - parTT), 0.0f), SNOWPACK);
            MELTWATER += melt;  SNOWPACK -= melt;
            const float refreeze = fminf(fmaxf(refrC * (parTT - Tt), 0.0f), MELTWATER);
            SNOWPACK += refreeze;  MELTWATER -= refreeze;
            const float tosoil = fmaxf(MELTWATER - parCWH * SNOWPACK, 0.0f);
            MELTWATER -= tosoil;

            // ---- soil bucket ----
            const float sw = fminf(__powf(SM * invFC, beta), 1.0f);   // SM > 0 always
            const float recharge = (RAIN + tosoil) * sw;
            SM += RAIN + tosoil - recharge;
            const float excess = fmaxf(SM - parFC, 0.0f);
            SM -= excess;
            const float ef = __powf(fminf(SM * invLPFC, 1.0f), betaet);
            const float ETact = fminf(SM, PETt * ef);
            SM = fmaxf(SM - ETact, NEARZERO);
            const float capillary = fminf(SLZ, parC * SLZ * (1.0f - fminf(SM * invFC, 1.0f)));
            SM  = fmaxf(SM + capillary, NEARZERO);
            SLZ = fmaxf(SLZ - capillary, NEARZERO);

            // ---- groundwater buckets ----
            SUZ += recharge + excess;
            const float PERC = fminf(SUZ, parPERC);
            SUZ -= PERC;
            const float Q0 = parK0 * fmaxf(SUZ - parUZL, 0.0f);
            SUZ -= Q0;
            const float Q1 = parK1 * SUZ;
            SUZ -= Q1;
            SLZ += PERC;
            const float gw = fminf(SLZ, gwmax);
            SLZ -= gw;
            const float Q2 = parK2 * SLZ;
            SLZ -= Q2;
            const float Qt = Q0 + Q1 + Q2;

            // ---- NMUL mean via wave32 lane-pair exchange ----
            const float Qm = 0.5f * (Qt + __shfl_xor(Qt, 1, 32));

            // ---- fused 15-tap gamma-UH routing (register ring buffer) ----
            float fl = w[0] * Qm;
            #pragma unroll
            for (int i = 0; i < LENF - 1; ++i) fl += w[i + 1] * h[i];
            #pragma unroll
            for (int i = LENF - 2; i > 0; --i) h[i] = h[i - 1];
            h[0] = Qm;

            if (writer) out[(size_t)(cb + tt) * N_GRID + g] = fl;
        }
        wait_async0();      // this chunk's prefetch complete
        __syncthreads();    // all waves done with old buffer / data visible
        buf ^= 1;
    }
#undef SX
#undef SD
}

extern "C" void kernel_launch(void* const* d_in, const int* in_sizes, int n_in,
                              void* d_out, int out_size, void* d_ws, size_t ws_size,
                              hipStream_t stream) {
    (void)in_sizes; (void)n_in; (void)d_ws; (void)ws_size; (void)out_size;
    const float* x_phy       = (const float*)d_in[0];
    const float* ac_all      = (const float*)d_in[1];
    // d_in[2] = elev_all: unused by the reference model
    const float* params_dy   = (const float*)d_in[3];
    const float* params_stat = (const float*)d_in[4];
    float* out = (float*)d_out;

    const int nblk = (N_GRID + GPB - 1) / GPB;   // 79 blocks of 256 threads
    hipLaunchKernelGGL(hbv_fused_kernel, dim3(nblk), dim3(BLOCK), 0, stream,
                       x_phy, ac_all, params_dy, params_stat, out);
}